// MaskBranch_1159641169978
// MI455X (gfx1250) — compile-verified
//
#include <hip/hip_runtime.h>

#define EPS 1e-6f

typedef float v2f __attribute__((ext_vector_type(2)));
typedef float v8f __attribute__((ext_vector_type(8)));

__device__ __forceinline__ void atomicMaxPos(float* addr, float v) {
    // all values are >= 0 -> IEEE bit pattern is monotone -> integer max works
    atomicMax((unsigned int*)addr, __float_as_uint(v));
}

__device__ __forceinline__ float waveReduceMax(float v) {
    #pragma unroll
    for (int m = 16; m > 0; m >>= 1)
        v = fmaxf(v, __shfl_xor(v, m, 32));
    return v;
}

// ---------------------------------------------------------------------------
// Stage 0: fused 4x4 stride-2 pad-1 conv + 2x2 maxpool (1024^2 -> 256^2).
// Never materializes the 512^2 conv output: max(maxpool(U)) == max(U).
// ---------------------------------------------------------------------------
__global__ void k_conv4_pool(const float* __restrict__ in, const float* __restrict__ w0p,
                             float* __restrict__ out, float* __restrict__ Mslot) {
    const int HI = 1024, HO = 256, B = 16;
    int idx = blockIdx.x * blockDim.x + threadIdx.x;
    int total = B * HO * HO;
    float pooled = 0.0f;
    if (idx < total) {
        int j = idx % HO;
        int t = idx / HO;
        int i = t % HO;
        int b = t / HO;
        const float* src = in + (size_t)b * HI * HI;
        float w0 = w0p[0];
        float x[6][6];
        #pragma unroll
        for (int r = 0; r < 6; ++r) {
            int gr = 4 * i - 1 + r;
            #pragma unroll
            for (int c = 0; c < 6; ++c) {
                int gc = 4 * j - 1 + c;
                x[r][c] = (gr >= 0 && gr < HI && gc >= 0 && gc < HI)
                            ? src[(size_t)gr * HI + gc] : 0.0f;
            }
        }
        float rs0[6], rs2[6];
        #pragma unroll
        for (int r = 0; r < 6; ++r) {
            rs0[r] = x[r][0] + x[r][1] + x[r][2] + x[r][3];
            rs2[r] = x[r][2] + x[r][3] + x[r][4] + x[r][5];
        }
        float s00 = rs0[0] + rs0[1] + rs0[2] + rs0[3];
        float s01 = rs2[0] + rs2[1] + rs2[2] + rs2[3];
        float s10 = rs0[2] + rs0[3] + rs0[4] + rs0[5];
        float s11 = rs2[2] + rs2[3] + rs2[4] + rs2[5];
        pooled = fmaxf(fmaxf(s00, s01), fmaxf(s10, s11)) * w0;
        out[idx] = pooled;
    }
    float m = waveReduceMax(pooled);
    if ((threadIdx.x & 31) == 0) atomicMaxPos(Mslot, m);
}

// in1 = (P/(M+eps)) / (M/(M+eps) + eps)   -- folded double normalization
__global__ void k_norm2(const float* __restrict__ src, const float* __restrict__ Mp,
                        float* __restrict__ dst, int n) {
    int i = blockIdx.x * blockDim.x + threadIdx.x;
    if (i < n) {
        float M = Mp[0];
        float d1 = M + EPS;
        float t = M / d1;
        float q = src[i] / d1;
        dst[i] = q / (t + EPS);
    }
}

__global__ void k_norm1(const float* __restrict__ src, const float* __restrict__ Mp,
                        float* __restrict__ dst, int n) {
    int i = blockIdx.x * blockDim.x + threadIdx.x;
    if (i < n) dst[i] = src[i] / (Mp[0] + EPS);
}

// ---------------------------------------------------------------------------
// 3x3 stride-2 pad-1 conv (small tensors only) — plain VALU, bandwidth-trivial.
// ---------------------------------------------------------------------------
__global__ void k_conv3_s2(const float* __restrict__ in, const float* __restrict__ w3p,
                           float* __restrict__ out, int Nin) {
    const int B = 16;
    int Nout = Nin >> 1;
    int idx = blockIdx.x * blockDim.x + threadIdx.x;
    int total = B * Nout * Nout;
    if (idx >= total) return;
    int j = idx % Nout;
    int t = idx / Nout;
    int i = t % Nout;
    int b = t / Nout;
    const float* src = in + (size_t)b * Nin * Nin;
    float s = 0.0f;
    #pragma unroll
    for (int di = 0; di < 3; ++di) {
        int gr = 2 * i - 1 + di;
        if (gr < 0 || gr >= Nin) continue;
        #pragma unroll
        for (int dj = 0; dj < 3; ++dj) {
            int gc = 2 * j - 1 + dj;
            if (gc >= 0 && gc < Nin) s += src[(size_t)gr * Nin + gc];
        }
    }
    out[idx] = s * w3p[0];
}

// ---------------------------------------------------------------------------
// 3x3 stride-1 pad-1 conv via the matrix pipe:
//   out(16x16) = V(16x20) * X(20x20) * (w * H(20x16))
// with banded 0/1 matrices (separable box filter), fifteen V_WMMA_F32_16X16X4_F32
// per tile. One wave per 16x16 tile, X and the intermediate Y staged in LDS.
// A-layout (16x4 f32):  lane = m + 16*(k/2), vgpr = k%2
// B-layout (4x16 f32):  lane = n + 16*(k/2), vgpr = k%2
// C/D layout (16x16):   vgpr r, lanes 0-15 -> M=r, lanes 16-31 -> M=r+8, N=lane%16
// ---------------------------------------------------------------------------
__global__ void __launch_bounds__(32)
k_conv3_wmma(const float* __restrict__ in, const float* __restrict__ w3p,
             float* __restrict__ out, float* __restrict__ Mslot, int N) {
    __shared__ float X[20][20];
    __shared__ float Y[16][20];

    const int lane = threadIdx.x;
    const int h = lane >> 4;      // lane half (selects K pair)
    const int j = lane & 15;      // M index (A) / N index (B,D)
    const int r0 = blockIdx.y * 16;
    const int c0 = blockIdx.x * 16;
    const int b  = blockIdx.z;
    const float* src = in + (size_t)b * N * N;

    // stage 20x20 input tile (origin at (r0-1, c0-1)), zero-padded
    for (int idx = lane; idx < 400; idx += 32) {
        int rr = idx / 20, cc = idx % 20;
        int gr = r0 - 1 + rr, gc = c0 - 1 + cc;
        float v = 0.0f;
        if (gr >= 0 && gr < N && gc >= 0 && gc < N) v = src[(size_t)gr * N + gc];
        X[rr][cc] = v;
    }
    __syncthreads();

    // vertical pass: Ya = V * X[:,0:16], Yb = V * X[:,16:20] (cols >=4 discarded)
    v8f ya = {}; v8f yb = {};
    #pragma unroll
    for (int q = 0; q < 5; ++q) {
        int k0 = 4 * q + 2 * h;
        int k1 = k0 + 1;
        v2f a;   // V[m][k] = 1 iff m <= k <= m+2
        a.x = (k0 >= j && k0 <= j + 2) ? 1.0f : 0.0f;
        a.y = (k1 >= j && k1 <= j + 2) ? 1.0f : 0.0f;
        v2f b0; b0.x = X[k0][j];  b0.y = X[k1][j];
        ya = __builtin_amdgcn_wmma_f32_16x16x4_f32(false, a, false, b0, (short)0, ya, false, false);
        int cj = 16 + j; if (cj > 19) cj = 19;     // clamped reads feed discarded D cols
        v2f b1; b1.x = X[k0][cj]; b1.y = X[k1][cj];
        yb = __builtin_amdgcn_wmma_f32_16x16x4_f32(false, a, false, b1, (short)0, yb, false, false);
    }

    // D-layout -> LDS (Y is 16 rows x 20 cols), then re-read in A-layout
    #pragma unroll
    for (int r = 0; r < 8; ++r) Y[r + 8 * h][j] = ya[r];
    if (j < 4) {
        #pragma unroll
        for (int r = 0; r < 8; ++r) Y[r + 8 * h][16 + j] = yb[r];
    }
    __syncthreads();

    // horizontal pass: D = Y * (w*H),  H[k][n] = 1 iff n <= k <= n+2
    float w = w3p[0];
    v8f d = {};
    #pragma unroll
    for (int q = 0; q < 5; ++q) {
        int k0 = 4 * q + 2 * h;
        int k1 = k0 + 1;
        v2f a; a.x = Y[j][k0]; a.y = Y[j][k1];
        v2f bb;
        int d0 = k0 - j, d1 = k1 - j;
        bb.x = (d0 >= 0 && d0 <= 2) ? w : 0.0f;
        bb.y = (d1 >= 0 && d1 <= 2) ? w : 0.0f;
        d = __builtin_amdgcn_wmma_f32_16x16x4_f32(false, a, false, bb, (short)0, d, false, false);
    }

    float* dst = out + (size_t)b * N * N;
    float mloc = 0.0f;
    #pragma unroll
    for (int r = 0; r < 8; ++r) {
        float v = d[r];
        dst[(size_t)(r0 + r + 8 * h) * N + (c0 + j)] = v;
        mloc = fmaxf(mloc, v);
    }
    mloc = waveReduceMax(mloc);
    if (lane == 0) atomicMaxPos(Mslot, mloc);
}

// ---------------------------------------------------------------------------
extern "C" void kernel_launch(void* const* d_in, const int* in_sizes, int n_in,
                              void* d_out, int out_size, void* d_ws, size_t ws_size,
                              hipStream_t stream) {
    (void)in_sizes; (void)n_in; (void)out_size; (void)ws_size;
    const float* m  = (const float*)d_in[0];
    const float* w0 = (const float*)d_in[1];   // 1/16
    const float* w3 = (const float*)d_in[2];   // 1/9
    float* out = (float*)d_out;

    char*  ws   = (char*)d_ws;
    float* M    = (float*)ws;                       // 16 max slots
    float* buf1 = (float*)(ws + 256);               // 16*256*256 f32
    float* buf2 = buf1 + (size_t)16 * 256 * 256;    // 16*256*256 f32

    hipMemsetAsync(M, 0, 16 * sizeof(float), stream);

    const int B = 16;
    float* o_in1  = out;
    float* o_out1 = o_in1  + (size_t)B * 256 * 256;
    float* o_in2  = o_out1 + (size_t)B * 256 * 256;
    float* o_out2 = o_in2  + (size_t)B * 256 * 256;
    float* o_in3  = o_out2 + (size_t)B * 128 * 128;
    float* o_out3 = o_in3  + (size_t)B * 128 * 128;
    float* o_in4  = o_out3 + (size_t)B * 64 * 64;
    float* o_out4 = o_in4  + (size_t)B * 64 * 64;
    float* o_out5 = o_out4 + (size_t)B * 32 * 32;

    // stage 0: fused conv4(s2) + maxpool, then folded double norm -> in1
    {
        int total = B * 256 * 256;
        k_conv4_pool<<<(total + 255) / 256, 256, 0, stream>>>(m, w0, buf1, &M[0]);
        k_norm2<<<(total + 255) / 256, 256, 0, stream>>>(buf1, &M[0], o_in1, total);
    }

    auto stage_s1 = [&](const float* inp, int N, float* dst, int slot) {
        dim3 g(N / 16, N / 16, B), blk(32);
        k_conv3_wmma<<<g, blk, 0, stream>>>(inp,  w3, buf1, &M[15], N);  // M[15] = dummy
        k_conv3_wmma<<<g, blk, 0, stream>>>(buf1, w3, buf2, &M[slot], N);
        int total = B * N * N;
        k_norm1<<<(total + 255) / 256, 256, 0, stream>>>(buf2, &M[slot], dst, total);
    };
    auto stage_s2 = [&](const float* inp, int Nin, float* dst, int slot) {
        int Nout = Nin / 2;
        int total = B * Nout * Nout;
        k_conv3_s2<<<(total + 255) / 256, 256, 0, stream>>>(inp, w3, buf1, Nin);
        dim3 g(Nout / 16, Nout / 16, B), blk(32);
        k_conv3_wmma<<<g, blk, 0, stream>>>(buf1, w3, buf2, &M[slot], Nout);
        k_norm1<<<(total + 255) / 256, 256, 0, stream>>>(buf2, &M[slot], dst, total);
    };

    stage_s1(o_in1,  256, o_out1, 1);   // out1
    stage_s1(o_out1, 256, o_in2,  2);   // in2
    stage_s2(o_in2,  256, o_out2, 3);   // out2 (s2 then s1)
    stage_s1(o_out2, 128, o_in3,  4);   // in3
    stage_s2(o_in3,  128, o_out3, 5);   // out3
    stage_s1(o_out3,  64, o_in4,  6);   // in4
    stage_s2(o_in4,   64, o_out4, 7);   // out4
    stage_s1(o_out4,  32, o_out5, 8);   // out5
}